// SVDpp_27608049779273
// MI455X (gfx1250) — compile-verified
//
#include <hip/hip_runtime.h>
#include <math.h>

// SVD++ forward for MI455X (gfx1250, wave32).
// Kernel 1: per-row CSR offsets via binary search on sorted segment_ids (no atomics -> deterministic).
// Kernel 2: one wave per 16-row tile.
//   Phase 1: ragged implicit gather (float2/lane, 4-deep unrolled for MLP) + explicit gathers,
//            Q = s_emb + y_u and P = p_emb staged in LDS (row stride 65 floats -> bank-conflict free).
//   Phase 2: D = Q x P^T via 16 chained V_WMMA_F32_16X16X4_F32 (K=64), diagonal extracted per the
//            documented 16x16 f32 C/D layout, bias added, stored.

typedef __attribute__((ext_vector_type(2))) float v2f;
typedef __attribute__((ext_vector_type(8))) float v8f;

#define GLOBAL_MEAN 3.82f
#define LDS_STRIDE 65              // 64 + 1 pad -> conflict-free column reads
#define WAVE_LDS (2 * 16 * LDS_STRIDE + 16)  // Q tile + P tile + 16 biases = 2096 floats
#define WAVES_PER_BLOCK 4

__global__ void svdpp_row_offsets(const int* __restrict__ seg,
                                  int* __restrict__ offs, int B, int T) {
    int t = blockIdx.x * blockDim.x + threadIdx.x;
    if (t > B) return;
    int lo = 0, hi = T;
    while (lo < hi) {                      // lower_bound: first i with seg[i] >= t
        int mid = (lo + hi) >> 1;
        if (seg[mid] < t) lo = mid + 1; else hi = mid;
    }
    offs[t] = lo;
}

__device__ __forceinline__ float pick8(v8f v, int i) {
    float r = v[0];
    if (i == 1) r = v[1];
    if (i == 2) r = v[2];
    if (i == 3) r = v[3];
    if (i == 4) r = v[4];
    if (i == 5) r = v[5];
    if (i == 6) r = v[6];
    if (i == 7) r = v[7];
    return r;
}

__global__ __launch_bounds__(32 * WAVES_PER_BLOCK)
void svdpp_fwd(const int* __restrict__ scientist_ids,
               const int* __restrict__ paper_ids,
               const int* __restrict__ flat_papers,
               const int* __restrict__ offs,
               const float* __restrict__ sfac,
               const float* __restrict__ pfac,
               const float* __restrict__ ifac,
               const float* __restrict__ sbias,
               const float* __restrict__ pbias,
               float* __restrict__ out) {
    const int lane = threadIdx.x & 31;
    const int wave = threadIdx.x >> 5;
    const int b0   = (blockIdx.x * WAVES_PER_BLOCK + wave) * 16;  // 16 rows per wave

    __shared__ float lds[WAVES_PER_BLOCK * WAVE_LDS];
    float* QL = lds + wave * WAVE_LDS;       // [16][65] : Q = s_emb + y_u
    float* PL = QL + 16 * LDS_STRIDE;        // [16][65] : P = p_emb
    float* BI = PL + 16 * LDS_STRIDE;        // [16]     : s_b + p_b + mean

    const v2f* sf2 = (const v2f*)sfac;       // 32 float2 per 64-dim row
    const v2f* pf2 = (const v2f*)pfac;
    const v2f* if2 = (const v2f*)ifac;

    // ---------- Phase 1: gathers + ragged segment sum (wave-uniform control flow) ----------
    for (int r = 0; r < 16; ++r) {
        const int b = b0 + r;
        const int s = offs[b];
        const int e = offs[b + 1];
        const int n = e - s;

        float ax = 0.f, ay = 0.f;
        int i = s;
        for (; i + 4 <= e; i += 4) {         // 4 independent 256B row reads in flight (L2-resident table)
            const int p0 = flat_papers[i + 0];
            const int p1 = flat_papers[i + 1];
            const int p2 = flat_papers[i + 2];
            const int p3 = flat_papers[i + 3];
            const v2f v0 = if2[p0 * 32 + lane];
            const v2f v1 = if2[p1 * 32 + lane];
            const v2f v2 = if2[p2 * 32 + lane];
            const v2f v3 = if2[p3 * 32 + lane];
            ax += (v0.x + v1.x) + (v2.x + v3.x);
            ay += (v0.y + v1.y) + (v2.y + v3.y);
        }
        for (; i < e; ++i) {
            const int p = flat_papers[i];
            const v2f v = if2[p * 32 + lane];
            ax += v.x; ay += v.y;
        }
        const float scale = (n > 0) ? rsqrtf((float)n) : 0.f;

        const int sid = scientist_ids[b];
        const int pid = paper_ids[b];
        const v2f se = sf2[sid * 32 + lane];
        const v2f pe = pf2[pid * 32 + lane];

        QL[r * LDS_STRIDE + 2 * lane + 0] = se.x + ax * scale;
        QL[r * LDS_STRIDE + 2 * lane + 1] = se.y + ay * scale;
        PL[r * LDS_STRIDE + 2 * lane + 0] = pe.x;
        PL[r * LDS_STRIDE + 2 * lane + 1] = pe.y;
        if (lane == 0)
            BI[r] = sbias[sid] + pbias[pid] + GLOBAL_MEAN;
    }
    // Wave-private LDS; same-wave DS ops are in-order -> no barrier needed.

    // ---------- Phase 2: D = Q x P^T with V_WMMA_F32_16X16X4_F32, K = 64 ----------
    // A 16x4 layout: lanes 0-15 hold M=lane (K=k0,k0+1 in the 2 VGPRs), lanes 16-31 K=k0+2,k0+3.
    // B 4x16 layout mirrors it with N=lane&15 -> identical LDS addressing for both operands.
    const int mm = lane & 15;
    const int kh = (lane >> 4) * 2;
    v8f acc = {};
#pragma unroll
    for (int k = 0; k < 16; ++k) {
        const int k0 = 4 * k + kh;
        v2f a, bb;
        a.x  = QL[mm * LDS_STRIDE + k0 + 0];
        a.y  = QL[mm * LDS_STRIDE + k0 + 1];
        bb.x = PL[mm * LDS_STRIDE + k0 + 0];
        bb.y = PL[mm * LDS_STRIDE + k0 + 1];
        // 8 args: (neg_a, A, neg_b, B, c_mod, C, reuse_a, reuse_b)
        acc = __builtin_amdgcn_wmma_f32_16x16x4_f32(
            false, a, false, bb, (short)0, acc, false, false);
    }

    // Diagonal of 16x16 f32 D: (i,i) for i<8 -> lane i, VGPR i; for i>=8 -> lane i+16, VGPR i-8.
    if (lane < 8) {
        const int m = lane;
        out[b0 + m] = pick8(acc, lane) + BI[m];
    } else if (lane >= 24) {
        const int m = lane - 16;
        out[b0 + m] = pick8(acc, lane - 24) + BI[m];
    }
}

extern "C" void kernel_launch(void* const* d_in, const int* in_sizes, int n_in,
                              void* d_out, int out_size, void* d_ws, size_t ws_size,
                              hipStream_t stream) {
    const int*   scientist_ids = (const int*)d_in[0];
    const int*   paper_ids     = (const int*)d_in[1];
    const int*   flat_papers   = (const int*)d_in[2];
    const int*   segment_ids   = (const int*)d_in[3];
    const float* sfac          = (const float*)d_in[4];
    const float* pfac          = (const float*)d_in[5];
    const float* ifac          = (const float*)d_in[6];
    const float* sbias         = (const float*)d_in[7];
    const float* pbias         = (const float*)d_in[8];
    float*       out           = (float*)d_out;

    const int B = in_sizes[0];   // 262144
    const int T = in_sizes[2];   // 5242880
    int* offs = (int*)d_ws;      // (B+1) ints of scratch

    svdpp_row_offsets<<<(B + 1 + 255) / 256, 256, 0, stream>>>(segment_ids, offs, B, T);

    // B is a multiple of 16*WAVES_PER_BLOCK (262144 / 64 = 4096).
    svdpp_fwd<<<B / (16 * WAVES_PER_BLOCK), 32 * WAVES_PER_BLOCK, 0, stream>>>(
        scientist_ids, paper_ids, flat_papers, offs,
        sfac, pfac, ifac, sbias, pbias, out);
}